// LTOCF_45784351375380
// MI455X (gfx1250) — compile-verified
//
#include <hip/hip_runtime.h>

// ---------------------------------------------------------------------------
// LightGCN on MI455X (gfx1250, wave32).
//  - SpMM layers: memory/atomic bound -> coalesced float4 gathers +
//    native f32 global atomics (relaxed, agent scope).
//  - Final batched dot: V_WMMA_F32_16X16X4_F32 (full fp32 matrix path),
//    diagonal extracted via LDS.
// ---------------------------------------------------------------------------

typedef __attribute__((ext_vector_type(2))) float v2f;
typedef __attribute__((ext_vector_type(8))) float v8f;

#define DEV_ATOMIC_ADD(p, v) \
    __hip_atomic_fetch_add((p), (v), __ATOMIC_RELAXED, __HIP_MEMORY_SCOPE_AGENT)

// deg[src] += 1 for src = concat(edge_u, edge_i + U): each undirected edge
// bumps both endpoints once.
__global__ void lgcn_degree(const int* __restrict__ eu, const int* __restrict__ ei,
                            float* __restrict__ deg, int E, int U) {
    int t = blockIdx.x * blockDim.x + threadIdx.x;
    if (t >= E) return;
    DEV_ATOMIC_ADD(&deg[eu[t]], 1.0f);
    DEV_ATOMIC_ADD(&deg[U + ei[t]], 1.0f);
}

// inv[n] = deg>0 ? 1/sqrt(deg) : 0   (in place; full-precision sqrt+div)
__global__ void lgcn_invsqrt(float* __restrict__ d, int N) {
    int t = blockIdx.x * blockDim.x + threadIdx.x;
    if (t >= N) return;
    float v = d[t];
    d[t] = (v > 0.0f) ? (1.0f / sqrtf(v)) : 0.0f;
}

// x0 = concat(user_emb, item_emb)
__global__ void lgcn_init_x0(const float* __restrict__ ue, const float* __restrict__ ie,
                             float* __restrict__ x, int U, int N) {
    long long t = (long long)blockIdx.x * blockDim.x + threadIdx.x;
    long long total = (long long)N * 64;
    if (t >= total) return;
    long long ubytes = (long long)U * 64;
    x[t] = (t < ubytes) ? ue[t] : ie[t - ubytes];
}

// xout[dst] += inv[src]*inv[dst] * xin[src]  over both edge directions.
// 16 threads per directed edge, float4 per thread -> each half-wave streams
// one contiguous 256B embedding row (coalesced), scatter via f32 atomics.
__global__ void lgcn_spmm(const int* __restrict__ eu, const int* __restrict__ ei,
                          const float* __restrict__ inv,
                          const float* __restrict__ xin, float* __restrict__ xout,
                          int E, int U) {
    long long t = (long long)blockIdx.x * blockDim.x + threadIdx.x;
    int edge = (int)(t >> 4);
    if (edge >= 2 * E) return;
    int q = ((int)t & 15) * 4;

    int s, d;
    if (edge < E) {
        s = eu[edge];
        d = U + ei[edge];
    } else {
        int e2 = edge - E;
        s = U + ei[e2];
        d = eu[e2];
    }
    float w = inv[s] * inv[d];
    const float4 v = *(const float4*)(xin + (size_t)s * 64 + q);
    float* o = xout + (size_t)d * 64 + q;
    DEV_ATOMIC_ADD(o + 0, w * v.x);
    DEV_ATOMIC_ADD(o + 1, w * v.y);
    DEV_ATOMIC_ADD(o + 2, w * v.z);
    DEV_ATOMIC_ADD(o + 3, w * v.w);
}

// Accumulate the 4096 user rows and 4096 item rows of the current embedding
// into small per-batch accumulators (running sum over the 5 layer outputs).
__global__ void lgcn_gather_acc(const int* __restrict__ users, const int* __restrict__ items,
                                const float* __restrict__ x,
                                float* __restrict__ uacc, float* __restrict__ iacc,
                                int U, int B) {
    int t = blockIdx.x * blockDim.x + threadIdx.x;
    int half = B * 64;
    if (t >= 2 * half) return;
    if (t < half) {
        int b = t >> 6, d = t & 63;
        uacc[t] += x[(size_t)users[b] * 64 + d];
    } else {
        int tt = t - half;
        int b = tt >> 6, d = tt & 63;
        iacc[tt] += x[((size_t)U + items[b]) * 64 + d];
    }
}

// gamma[b] = dot(uacc[b,:], iacc[b,:]) / 25 via V_WMMA_F32_16X16X4_F32.
// One wave handles a 16-row tile: C = U(16x64) x I^T(64x16), take diagonal.
// fp32 A/B operands: 2 per lane, K = 2*(lane>=16) + vgpr (symmetric for A/B,
// and we only consume m==n==lane&15 so the A/B row is the same row index).
__global__ __launch_bounds__(256) void lgcn_dot_wmma(const float* __restrict__ uacc,
                                                     const float* __restrict__ iacc,
                                                     float* __restrict__ out, int B) {
    __shared__ float tile[8][16 * 16];
    const int lane = threadIdx.x & 31;
    const int wave = threadIdx.x >> 5;
    const int tile0 = (blockIdx.x * 8 + wave) * 16;

    const int m = lane & 15;                 // A row == B column we feed
    const int khalf = (lane >> 4) << 1;      // 0 for lanes 0-15, 2 for 16-31
    int row = tile0 + m;
    if (row > B - 1) row = B - 1;            // clamp, keep EXEC all-ones
    const float* __restrict__ ur = uacc + (size_t)row * 64;
    const float* __restrict__ ir = iacc + (size_t)row * 64;

    v8f c = {};
#pragma unroll
    for (int kb = 0; kb < 64; kb += 4) {
        int k0 = kb + khalf;
        v2f a; a.x = ur[k0]; a.y = ur[k0 + 1];
        v2f b; b.x = ir[k0]; b.y = ir[k0 + 1];
        c = __builtin_amdgcn_wmma_f32_16x16x4_f32(false, a, false, b,
                                                  (short)0, c, false, false);
    }

    // C/D layout: VGPR r -> M = r + 8*(lane>=16), N = lane&15.
    float* lds = &tile[wave][0];
    const int n = lane & 15;
    const int mbase = (lane >> 4) ? 8 : 0;
#pragma unroll
    for (int r = 0; r < 8; ++r)
        lds[(mbase + r) * 16 + n] = c[r];
    __syncthreads();

    if (lane < 16) {
        int b = tile0 + lane;
        if (b < B)
            out[b] = lds[lane * 16 + lane] * (1.0f / 25.0f);
    }
}

extern "C" void kernel_launch(void* const* d_in, const int* in_sizes, int n_in,
                              void* d_out, int out_size, void* d_ws, size_t ws_size,
                              hipStream_t stream) {
    const int*   users = (const int*)d_in[0];
    const int*   items = (const int*)d_in[1];
    const int*   eu    = (const int*)d_in[2];
    const int*   ei    = (const int*)d_in[3];
    const float* uemb  = (const float*)d_in[4];
    const float* iemb  = (const float*)d_in[5];
    float*       out   = (float*)d_out;

    const int B = in_sizes[0];           // 4096
    const int E = in_sizes[2];           // 600000
    const int U = in_sizes[4] / 64;      // 100000
    const int I = in_sizes[5] / 64;      // 50000
    const int N = U + I;                 // 150000

    // Workspace carve-up (~79.5 MB total)
    char* ws = (char*)d_ws;
    size_t off = 0;
    auto carve = [&](size_t bytes) -> float* {
        float* p = (float*)(ws + off);
        off += (bytes + 255) & ~(size_t)255;
        return p;
    };
    float* inv    = carve((size_t)N * 4);            // deg, then inv_sqrt in place
    float* e_cur  = carve((size_t)N * 64 * 4);
    float* e_next = carve((size_t)N * 64 * 4);
    float* uacc   = carve((size_t)B * 64 * 4);
    float* iacc   = carve((size_t)B * 64 * 4);

    const int BS = 256;

    hipMemsetAsync(inv,  0, (size_t)N * 4, stream);
    hipMemsetAsync(uacc, 0, (size_t)B * 64 * 4, stream);
    hipMemsetAsync(iacc, 0, (size_t)B * 64 * 4, stream);

    {   // x0
        long long total = (long long)N * 64;
        lgcn_init_x0<<<(int)((total + BS - 1) / BS), BS, 0, stream>>>(uemb, iemb, e_cur, U, N);
    }
    lgcn_degree<<<(E + BS - 1) / BS, BS, 0, stream>>>(eu, ei, inv, E, U);
    lgcn_invsqrt<<<(N + BS - 1) / BS, BS, 0, stream>>>(inv, N);

    const int gacc_blocks = (2 * B * 64 + BS - 1) / BS;
    lgcn_gather_acc<<<gacc_blocks, BS, 0, stream>>>(users, items, e_cur, uacc, iacc, U, B);

    const long long spmm_threads = (long long)2 * E * 16;
    const int spmm_blocks = (int)((spmm_threads + BS - 1) / BS);
    for (int layer = 0; layer < 4; ++layer) {
        hipMemsetAsync(e_next, 0, (size_t)N * 64 * 4, stream);
        lgcn_spmm<<<spmm_blocks, BS, 0, stream>>>(eu, ei, inv, e_cur, e_next, E, U);
        lgcn_gather_acc<<<gacc_blocks, BS, 0, stream>>>(users, items, e_next, uacc, iacc, U, B);
        float* tmp = e_cur; e_cur = e_next; e_next = tmp;
    }

    // Final batched dot-diagonal on the fp32 WMMA path.
    const int tiles = (B + 15) / 16;
    const int dot_blocks = (tiles + 7) / 8;
    lgcn_dot_wmma<<<dot_blocks, 256, 0, stream>>>(uacc, iacc, out, B);
}